// BipolarSAE_86715389706467
// MI455X (gfx1250) — compile-verified
//
#include <hip/hip_runtime.h>
#include <stdint.h>

// ---------------------------------------------------------------------------
// TopK SAE forward for MI455X (gfx1250, wave32).
//   d_in: x[32768,768] f32, W_enc[4096,768], b_enc[4096], W_dec[768,4096], b_dec[768]
//   d_out: recon[32768,768] f32 || f[32768,4096] f32
//
// Roofline: mandatory HBM ~1.25 GB (x read, pre_acts writeback, f write,
// recon write) => ~54us @23.3TB/s. Encoder is a 128x128 LDS-tiled f32 WMMA
// GEMM (A/B cache traffic ~6.4GB, well under HBM-bound time). Encoder+topk
// are chunked over 8192-token slices so the 128MB pre_acts slice is re-read
// from the 192MB L2 instead of HBM. Decoder exploits K=32 sparsity.
// ---------------------------------------------------------------------------

#define N_TOK   32768
#define D_IN    768
#define D_OUT   4096
#define TOPK    32
#define TCHUNK  8192          // tokens per encoder/topk slice (128MB pre_acts < L2)

typedef __attribute__((ext_vector_type(2))) float v2f;
typedef __attribute__((ext_vector_type(8))) float v8f;

// ---------------------------------------------------------------------------
// Kernel 0: W_dec [768,4096] -> W_decT [4096,768]
// ---------------------------------------------------------------------------
__global__ void sae_transpose_wdec(const float* __restrict__ Wd,
                                   float* __restrict__ WdT) {
    __shared__ float tile[32][33];
    const int f0 = blockIdx.x * 32;
    const int d0 = blockIdx.y * 32;
    const int tx = threadIdx.x & 31;
    const int ty = threadIdx.x >> 5;
#pragma unroll
    for (int i = 0; i < 32; i += 8)
        tile[ty + i][tx] = Wd[(size_t)(d0 + ty + i) * D_OUT + f0 + tx];
    __syncthreads();
#pragma unroll
    for (int i = 0; i < 32; i += 8)
        WdT[(size_t)(f0 + ty + i) * D_IN + d0 + tx] = tile[tx][ty + i];
}

// ---------------------------------------------------------------------------
// Kernel 1: encoder GEMM, 128x128 tile per block, V_WMMA_F32_16X16X4_F32.
// 256 threads = 8 waves; wave grid 4x2 over (token,feature) subtiles, each
// wave owns a 2x4 array of 16x16 accumulators (A reused 4x, B reused 2x).
// K staged in LDS chunks of 32 with pitch 36 (gcd(36,64)=4, 36*t mod 64
// distinct for t=0..15 -> conflict-free 64-bit fragment reads).
// ---------------------------------------------------------------------------
#define MT     128
#define NT     128
#define KC     32
#define LPITCH 36

__global__ void sae_encoder_wmma(const float* __restrict__ x,
                                 const float* __restrict__ W_enc,
                                 const float* __restrict__ b_enc,
                                 float* __restrict__ f_out,
                                 int tok_base) {
    __shared__ float xs[MT * LPITCH];           // 18,432 B
    __shared__ float wsh[NT * LPITCH];          // 18,432 B

    const int tid   = threadIdx.x;
    const int lane  = tid & 31;
    const int wave  = tid >> 5;
    const int l16   = lane & 15;
    const int lhalf = lane >> 4;                // 0: K0/K1, 1: K2/K3 (ISA A/B layout)
    const int tbase = tok_base + blockIdx.y * MT;
    const int fbase = blockIdx.x * NT;
    const int wt    = (wave & 3) * 32;          // wave's token offset (2 subtiles)
    const int wf    = (wave >> 2) * 64;         // wave's feature offset (4 subtiles)

    v8f acc[2][4] = {};

    for (int kb = 0; kb < D_IN; kb += KC) {
        // Cooperative float4 staging of x[128xKC] and W_enc[128xKC] chunks.
        for (int i = tid; i < MT * (KC / 4); i += 256) {
            const int r = i >> 3, c4 = i & 7;
            *(float4*)&xs[r * LPITCH + c4 * 4] =
                *(const float4*)&x[(size_t)(tbase + r) * D_IN + kb + c4 * 4];
        }
        for (int i = tid; i < NT * (KC / 4); i += 256) {
            const int r = i >> 3, c4 = i & 7;
            *(float4*)&wsh[r * LPITCH + c4 * 4] =
                *(const float4*)&W_enc[(size_t)(fbase + r) * D_IN + kb + c4 * 4];
        }
        __syncthreads();

#pragma unroll
        for (int ko = 0; ko < KC; ko += 4) {
            const int o = ko + lhalf * 2;
            v2f a0 = *(const v2f*)&xs[(wt + l16) * LPITCH + o];
            v2f a1 = *(const v2f*)&xs[(wt + 16 + l16) * LPITCH + o];
            v2f b0 = *(const v2f*)&wsh[(wf + l16) * LPITCH + o];
            v2f b1 = *(const v2f*)&wsh[(wf + 16 + l16) * LPITCH + o];
            v2f b2 = *(const v2f*)&wsh[(wf + 32 + l16) * LPITCH + o];
            v2f b3 = *(const v2f*)&wsh[(wf + 48 + l16) * LPITCH + o];
#define WM(A, B, C) __builtin_amdgcn_wmma_f32_16x16x4_f32(false, A, false, B, (short)0, C, false, false)
            acc[0][0] = WM(a0, b0, acc[0][0]);
            acc[0][1] = WM(a0, b1, acc[0][1]);
            acc[0][2] = WM(a0, b2, acc[0][2]);
            acc[0][3] = WM(a0, b3, acc[0][3]);
            acc[1][0] = WM(a1, b0, acc[1][0]);
            acc[1][1] = WM(a1, b1, acc[1][1]);
            acc[1][2] = WM(a1, b2, acc[1][2]);
            acc[1][3] = WM(a1, b3, acc[1][3]);
#undef WM
        }
        __syncthreads();
    }

    // Epilogue: + b_enc, store pre_acts. C/D layout: VGPR r -> M = r+8*lhalf, N = l16.
#pragma unroll
    for (int j = 0; j < 4; ++j) {
        const int feat = fbase + wf + j * 16 + l16;
        const float bias = b_enc[feat];
#pragma unroll
        for (int i = 0; i < 2; ++i) {
#pragma unroll
            for (int r = 0; r < 8; ++r) {
                const int tok = tbase + wt + i * 16 + r + 8 * lhalf;
                f_out[(size_t)tok * D_OUT + feat] = acc[i][j][r] + bias;
            }
        }
    }
}

// ---------------------------------------------------------------------------
// Kernel 2: exact top-32 by |value| via 4-pass radix select on magnitude
// bits; masks f in place and compacts (idx,val) pairs to workspace.
// ---------------------------------------------------------------------------
__global__ void sae_topk_mask(float* __restrict__ f_out,
                              float* __restrict__ ws_val,
                              int*   __restrict__ ws_idx,
                              int tok_base) {
    __shared__ uint32_t bits[D_OUT];
    __shared__ uint32_t hist[256];
    __shared__ uint32_t sh_digit, sh_rem, sh_tie, sh_nsel;

    const int tid = threadIdx.x;
    const size_t token = (size_t)tok_base + blockIdx.x;
    const size_t row = token * D_OUT;

#pragma unroll
    for (int t = 0; t < 16; ++t) {
        const int j = tid + t * 256;
        bits[j] = __float_as_uint(f_out[row + j]);
    }
    if (tid == 0) { sh_rem = TOPK; sh_tie = 0; sh_nsel = 0; }
    if (tid < TOPK) {
        ws_val[token * TOPK + tid] = 0.0f;
        ws_idx[token * TOPK + tid] = 0;
    }
    __syncthreads();

    uint32_t prefHigh = 0;
    for (int pass = 0; pass < 4; ++pass) {
        const int shift = 24 - pass * 8;
        hist[tid] = 0;
        __syncthreads();
#pragma unroll
        for (int t = 0; t < 16; ++t) {
            const uint32_t m  = bits[tid + t * 256] & 0x7fffffffu;
            const uint32_t hi = (pass == 0) ? 0u : (m >> (shift + 8));
            if (hi == prefHigh) atomicAdd(&hist[(m >> shift) & 255u], 1u);
        }
        __syncthreads();
        if (tid == 0) {
            uint32_t rem = sh_rem, cum = 0;
            int b = 255;
            for (; b > 0; --b) { cum += hist[b]; if (cum >= rem) break; }
            if (cum < rem) cum += hist[0];
            sh_digit = (uint32_t)b;
            sh_rem   = rem - (cum - hist[b]);
        }
        __syncthreads();
        prefHigh = (prefHigh << 8) | sh_digit;
        __syncthreads();
    }
    const uint32_t thresh = prefHigh;
    const uint32_t rties  = sh_rem;

#pragma unroll
    for (int t = 0; t < 16; ++t) {
        const int j = tid + t * 256;
        const uint32_t ub = bits[j];
        const uint32_t m  = ub & 0x7fffffffu;
        bool keep = (m > thresh);
        if (m == thresh) keep = (atomicAdd(&sh_tie, 1u) < rties);
        const float v = __uint_as_float(ub);
        f_out[row + j] = keep ? v : 0.0f;
        if (keep) {
            const uint32_t p = atomicAdd(&sh_nsel, 1u);
            if (p < TOPK) {
                ws_val[token * TOPK + p] = v;
                ws_idx[token * TOPK + p] = j;
            }
        }
    }
}

// ---------------------------------------------------------------------------
// Kernel 3: sparse decoder. recon[n,d] = b_dec[d] + sum_k val_k*W_decT[idx_k][d]
// ---------------------------------------------------------------------------
__global__ void sae_decode_sparse(const float* __restrict__ WdT,
                                  const float* __restrict__ b_dec,
                                  const float* __restrict__ ws_val,
                                  const int*   __restrict__ ws_idx,
                                  float* __restrict__ recon) {
    __shared__ float sv[TOPK];
    __shared__ int   si[TOPK];
    const int tid = threadIdx.x;
    const size_t n = blockIdx.x;
    if (tid < TOPK) {
        sv[tid] = ws_val[n * TOPK + tid];
        si[tid] = ws_idx[n * TOPK + tid];
    }
    __syncthreads();

    float a0 = b_dec[tid], a1 = b_dec[tid + 256], a2 = b_dec[tid + 512];
#pragma unroll
    for (int k = 0; k < TOPK; ++k) {
        const float* wr = WdT + (size_t)si[k] * D_IN;
        const float  v  = sv[k];
        a0 += v * wr[tid];
        a1 += v * wr[tid + 256];
        a2 += v * wr[tid + 512];
    }
    recon[n * D_IN + tid]       = a0;
    recon[n * D_IN + tid + 256] = a1;
    recon[n * D_IN + tid + 512] = a2;
}

// ---------------------------------------------------------------------------
extern "C" void kernel_launch(void* const* d_in, const int* in_sizes, int n_in,
                              void* d_out, int out_size, void* d_ws, size_t ws_size,
                              hipStream_t stream) {
    const float* x     = (const float*)d_in[0];
    const float* W_enc = (const float*)d_in[1];
    const float* b_enc = (const float*)d_in[2];
    const float* W_dec = (const float*)d_in[3];
    const float* b_dec = (const float*)d_in[4];

    float* recon = (float*)d_out;                    // [32768, 768]
    float* f_out = recon + (size_t)N_TOK * D_IN;     // [32768, 4096]

    // Workspace: W_decT (12.6MB) | vals (4MB) | idx (4MB)
    float* WdT    = (float*)d_ws;
    float* ws_val = WdT + (size_t)D_OUT * D_IN;
    int*   ws_idx = (int*)(ws_val + (size_t)N_TOK * TOPK);

    sae_transpose_wdec<<<dim3(D_OUT / 32, D_IN / 32), 256, 0, stream>>>(W_dec, WdT);

    // Chunk encoder+topk so each 128MB pre_acts slice is re-read from L2.
    for (int c = 0; c < N_TOK / TCHUNK; ++c) {
        const int tb = c * TCHUNK;
        sae_encoder_wmma<<<dim3(D_OUT / NT, TCHUNK / MT), 256, 0, stream>>>(
            x, W_enc, b_enc, f_out, tb);
        sae_topk_mask<<<TCHUNK, 256, 0, stream>>>(f_out, ws_val, ws_idx, tb);
    }

    sae_decode_sparse<<<N_TOK, 256, 0, stream>>>(WdT, b_dec, ws_val, ws_idx, recon);
}